// FilterDetections_62508954026450
// MI455X (gfx1250) — compile-verified
//
#include <hip/hip_runtime.h>
#include <stdint.h>

#define BATCH   8
#define NANCH   49104
#define NCLS    20
#define MAXDET  300
#define NEGV    (-1e9f)
#define HALF_NEG (-5e8f)
#define THRESH  0.05f
#define IOU_T   0.5f
#define BLK1    1024
#define NW1     (BLK1 / 32)
#define BLK2    512
#define NW2     (BLK2 / 32)
#define NCAND   (NCLS * MAXDET)   /* 6000 */
#define FLT_LOW (-3.402823e38f)

typedef unsigned u32x4 __attribute__((ext_vector_type(4)));
typedef unsigned u32x8 __attribute__((ext_vector_type(8)));

// ---- CDNA5 async global->LDS copy path (ASYNCcnt) ----------------------
__device__ __forceinline__ void async_lds_b128(unsigned lds_off,
                                               unsigned long long gaddr) {
  asm volatile("global_load_async_to_lds_b128 %0, %1, off"
               :: "v"(lds_off), "v"(gaddr) : "memory");
}
__device__ __forceinline__ void wait_async_zero() {
  asm volatile("s_wait_asynccnt 0x0" ::: "memory");
}

// ---- CDNA5 Tensor Data Mover: 1-D tile, global -> LDS (TENSORcnt) ------
// D# per CDNA5 ISA: group0 {count=1, lds_addr, global_addr,
// type=2}; group1 {data_size=4B, tensor_dim0=N, tile_dim0=N}.
__device__ __forceinline__ void tdm_load_1d_f32(unsigned lds_addr,
                                                const float* gptr,
                                                unsigned nelem) {
  unsigned long long ga = (unsigned long long)(uintptr_t)gptr;
  u32x4 g0;
  g0.x = 1u;                                   // count=1, user descriptor
  g0.y = lds_addr;                             // lds_addr [63:32]
  g0.z = (unsigned)ga;                         // global_addr [95:64]
  g0.w = (unsigned)(ga >> 32) | (2u << 30);    // global_addr[56:32] | type=2
  u32x8 g1;
  g1.s0 = (2u << 16);                          // data_size=4B, mask=0
  g1.s1 = (nelem & 0xFFFFu) << 16;             // tensor_dim0[15:0]
  g1.s2 = (nelem >> 16) | (1u << 16);          // tensor_dim0[31:16], tensor_dim1=1
  g1.s3 = (nelem & 0xFFFFu) << 16;             // tile_dim0
  g1.s4 = 0u;                                  // tile_dim1/2 unused
  g1.s5 = nelem;                               // tensor_dim0_stride[31:0]
  g1.s6 = 0u;
  g1.s7 = 0u;
  asm volatile("tensor_load_to_lds %0, %1" :: "s"(g0), "s"(g1) : "memory");
}

// ---- wave32 + LDS block argmax with first-index tie-break --------------
template <int NW>
__device__ __forceinline__ void block_argmax(float v, int i, float* rv, int* ri,
                                             float* bv, int* bi, int tid,
                                             float& outv, int& outi) {
  const int lane = tid & 31;
  const int w = tid >> 5;
#pragma unroll
  for (int off = 16; off > 0; off >>= 1) {
    float ov = __shfl_down(v, off, 32);
    int   oi = __shfl_down(i, off, 32);
    if (ov > v || (ov == v && oi < i)) { v = ov; i = oi; }
  }
  if (lane == 0) { rv[w] = v; ri[w] = i; }
  __syncthreads();
  if (w == 0) {
    float vv = (lane < NW) ? rv[lane] : FLT_LOW;
    int   ii = (lane < NW) ? ri[lane] : 0x7fffffff;
#pragma unroll
    for (int off = 16; off > 0; off >>= 1) {
      float ov = __shfl_down(vv, off, 32);
      int   oi = __shfl_down(ii, off, 32);
      if (ov > vv || (ov == vv && oi < ii)) { vv = ov; ii = oi; }
    }
    if (lane == 0) { *bv = vv; *bi = ii; }
  }
  __syncthreads();
  outv = *bv;
  outi = *bi;
}

// ---- kernel 1: [B,N,C] -> [B,C,N] so per-class scores are contiguous ---
extern "C" __global__ __launch_bounds__(256)
void transpose_cls(const float* __restrict__ cls, float* __restrict__ clsT) {
  long long g = (long long)blockIdx.x * 256 + threadIdx.x;
  if (g >= (long long)BATCH * NANCH) return;
  const int b = (int)(g / NANCH);
  const int j = (int)(g % NANCH);
  const float* src = cls + ((size_t)b * NANCH + j) * NCLS;
  float vals[NCLS];
#pragma unroll
  for (int c = 0; c < NCLS; ++c) vals[c] = src[c];
#pragma unroll
  for (int c = 0; c < NCLS; ++c)
    clsT[((size_t)b * NCLS + c) * NANCH + j] = vals[c];
}

// ---- kernel 2: greedy NMS, one workgroup per (batch, class) ------------
// Entire score vector (49104 f32 = 192KB) lives in LDS (CDNA5: 320KB/WGP).
extern "C" __global__ __launch_bounds__(BLK1)
void nms_per_class(const float* __restrict__ boxes,
                   const float* __restrict__ clsT,
                   float* __restrict__ cand_sc,
                   int*   __restrict__ cand_idx) {
  extern __shared__ float smem[];
  float* s  = smem;                    // NANCH scores
  float* rv = smem + NANCH;            // NW1 wave maxima
  int*   ri = (int*)(rv + NW1);        // NW1 wave argmax
  float* bv = (float*)(ri + NW1);      // broadcast value
  int*   bi = (int*)(bv + 1);          // broadcast index

  const int tid = threadIdx.x;
  const int b = blockIdx.x / NCLS;
  const int c = blockIdx.x % NCLS;
  const float* bx  = boxes + (size_t)b * NANCH * 4;
  const float* scp = clsT + ((size_t)b * NCLS + c) * NANCH;

  // TDM: one DMA moves the whole 192KB score vector into LDS (wave 0 only;
  // TDM ignores EXEC, so every issuing wave would start a duplicate DMA).
  const unsigned sbase = (unsigned)(uintptr_t)s;
  if (tid < 32) {
    tdm_load_1d_f32(sbase, scp, NANCH);
    __builtin_amdgcn_s_wait_tensorcnt(0);
  }
  __syncthreads();

  // threshold + initial argmax (each thread owns j = tid + k*BLK1 slots)
  float lv = FLT_LOW; int li = 0x7fffffff;
  for (int j = tid; j < NANCH; j += BLK1) {
    float v = s[j];
    v = (v > THRESH) ? v : NEGV;
    s[j] = v;
    if (v > lv) { lv = v; li = j; }
  }
  float bestv; int besti;
  block_argmax<NW1>(lv, li, rv, ri, bv, bi, tid, bestv, besti);

  const size_t cbase = (size_t)blockIdx.x * MAXDET;
  for (int it = 0; it < MAXDET; ++it) {
    const bool valid = bestv > HALF_NEG;          // uniform (broadcast)
    if (tid == 0) {
      cand_sc[cbase + it]  = valid ? bestv : NEGV;
      cand_idx[cbase + it] = besti;
    }
    if (!valid) {                                  // all suppressed: ref yields idx 0, NEG
      for (int k = it + 1 + tid; k < MAXDET; k += BLK1) {
        cand_sc[cbase + k]  = NEGV;
        cand_idx[cbase + k] = besti;
      }
      break;
    }
    if (it == MAXDET - 1) break;

    // fused suppression (IoU vs best) + next argmax, one LDS sweep.
    // Dead entries (already NEG) skip the box load + IoU entirely.
    const int i = besti;
    const float4 p = *(const float4*)(bx + 4 * (size_t)i);
    const float pa = (p.z - p.x) * (p.w - p.y);
    lv = FLT_LOW; li = 0x7fffffff;
    for (int j = tid; j < NANCH; j += BLK1) {
      float v = s[j];
      if (v > HALF_NEG) {
        const float4 q = *(const float4*)(bx + 4 * (size_t)j);
        const float lx = fmaxf(p.x, q.x);
        const float ly = fmaxf(p.y, q.y);
        const float rx = fminf(p.z, q.z);
        const float ry = fminf(p.w, q.w);
        const float iw = fmaxf(rx - lx, 0.0f);
        const float ih = fmaxf(ry - ly, 0.0f);
        const float inter = iw * ih;
        const float qa = (q.z - q.x) * (q.w - q.y);
        const float iou = inter / (qa + pa - inter + 1e-9f);
        if (iou > IOU_T || j == i) { v = NEGV; s[j] = NEGV; }
      }
      if (v > lv) { lv = v; li = j; }
    }
    block_argmax<NW1>(lv, li, rv, ri, bv, bi, tid, bestv, besti);
  }
}

// ---- kernel 3: per-batch stable top-300 of 6000 + gather/emit ----------
extern "C" __global__ __launch_bounds__(BLK2)
void topk_emit(const float* __restrict__ boxes,
               const float* __restrict__ b3d,
               const float* __restrict__ cand_sc,
               const int*   __restrict__ cand_idx,
               float* __restrict__ out) {
  __shared__ float s2[NCAND];
  __shared__ float rv[NW2];
  __shared__ int   ri[NW2];
  __shared__ float bvs;
  __shared__ int   bis;
  __shared__ int   self_f[MAXDET];
  __shared__ float self_v[MAXDET];

  const int tid = threadIdx.x;
  const int b = blockIdx.x;

  // async global->LDS (ASYNCcnt) staging of the candidate scores
  const unsigned sbase = (unsigned)(uintptr_t)s2;
  const float* src = cand_sc + (size_t)b * NCAND;
  for (int q = tid; q < NCAND / 4; q += BLK2)
    async_lds_b128(sbase + 16u * (unsigned)q,
                   (unsigned long long)(uintptr_t)(src + 4 * (size_t)q));
  wait_async_zero();
  __syncthreads();

  // 300 sequential argmaxes == lax.top_k order (ties -> lowest flat index)
  for (int it = 0; it < MAXDET; ++it) {
    float lv = FLT_LOW; int li = 0x7fffffff;
    for (int f = tid; f < NCAND; f += BLK2) {
      float v = s2[f];
      if (v > lv) { lv = v; li = f; }
    }
    float bvv; int bii;
    block_argmax<NW2>(lv, li, rv, ri, &bvs, &bis, tid, bvv, bii);
    if (tid == 0) { self_f[it] = bii; self_v[it] = bvv; s2[bii] = FLT_LOW; }
    __syncthreads();
  }

  const float* bxb = boxes + (size_t)b * NANCH * 4;
  const float* b3b = b3d   + (size_t)b * NANCH * 16;
  float* out_boxes = out;                                       // [B,300,4]
  float* out_b3d   = out + (size_t)BATCH * MAXDET * 4;          // [B,300,16]
  float* out_sc    = out + (size_t)BATCH * MAXDET * 20;         // [B,300]
  float* out_lab   = out_sc + (size_t)BATCH * MAXDET;           // [B,300]

  for (int k = tid; k < MAXDET; k += BLK2) {
    const int f = self_f[k];
    const float v = self_v[k];
    const bool ok = v > HALF_NEG;
    const int c  = f / MAXDET;
    const int it = f - c * MAXDET;
    const int anchor = cand_idx[((size_t)b * NCLS + c) * MAXDET + it];
    out_sc[(size_t)b * MAXDET + k]  = ok ? v : -1.0f;
    out_lab[(size_t)b * MAXDET + k] = ok ? (float)c : -1.0f;
    float4 bb;
    if (ok) bb = *(const float4*)(bxb + 4 * (size_t)anchor);
    else { bb.x = -1.0f; bb.y = -1.0f; bb.z = -1.0f; bb.w = -1.0f; }
    *(float4*)(out_boxes + ((size_t)b * MAXDET + k) * 4) = bb;
    const float* s3 = b3b + 16 * (size_t)anchor;
    float* d3 = out_b3d + ((size_t)b * MAXDET + k) * 16;
#pragma unroll
    for (int q2 = 0; q2 < 16; ++q2) d3[q2] = ok ? s3[q2] : -1.0f;
  }
}

extern "C" void kernel_launch(void* const* d_in, const int* in_sizes, int n_in,
                              void* d_out, int out_size, void* d_ws, size_t ws_size,
                              hipStream_t stream) {
  (void)in_sizes; (void)n_in; (void)out_size; (void)ws_size;
  const float* boxes = (const float*)d_in[0];
  const float* b3d   = (const float*)d_in[1];
  const float* cls   = (const float*)d_in[2];
  float* out = (float*)d_out;

  // workspace: cand_sc [B*C*300] f32 | cand_idx [B*C*300] i32 | clsT [B*C*N] f32
  float* cand_sc  = (float*)d_ws;
  int*   cand_idx = (int*)(cand_sc + (size_t)BATCH * NCLS * MAXDET);
  float* clsT     = (float*)(cand_idx + (size_t)BATCH * NCLS * MAXDET);

  {
    long long total = (long long)BATCH * NANCH;
    int grid = (int)((total + 255) / 256);
    transpose_cls<<<grid, 256, 0, stream>>>(cls, clsT);
  }
  {
    size_t smem1 = ((size_t)NANCH + 2 * NW1 + 2) * sizeof(float);
    nms_per_class<<<BATCH * NCLS, BLK1, smem1, stream>>>(boxes, clsT, cand_sc, cand_idx);
  }
  topk_emit<<<BATCH, BLK2, 0, stream>>>(boxes, b3d, cand_sc, cand_idx, out);
}